// GCN_19164144075571
// MI455X (gfx1250) — compile-verified
//
#include <hip/hip_runtime.h>
#include <math.h>

typedef __attribute__((ext_vector_type(2))) float v2f;
typedef __attribute__((ext_vector_type(8))) float v8f;

#define GN 100000
#define GE 1600000
#define F_IN 22
#define H1F 128
#define H2F 256
#define NCLS 4

// ---------------------------------------------------------------------------
// Dense GEMM via V_WMMA_F32_16X16X4_F32. One wave computes one 16x16 tile of
// C = A[M,K] @ B[K,NC]. M and NC must be multiples of 16 (M=100000=16*6250,
// NC=128/256). K handled in steps of 4 with a guarded tail (K=22 case).
// A-frag layout (ISA 7.12.2, 32-bit A 16x4): lane%16 = row, (lane>>4)*2 = k
// pair base, VGPR0=k0, VGPR1=k0+1. B symmetric (lane%16 = col).
// C/D: VGPR r holds M = r + 8*(lane>>4), N = lane%16.
// ---------------------------------------------------------------------------
__global__ void gemm_wmma_f32(const float* __restrict__ A,
                              const float* __restrict__ B,
                              float* __restrict__ Cout,
                              int M, int K, int NC) {
    int wave = (int)((blockIdx.x * (long long)blockDim.x + threadIdx.x) >> 5);
    int lane = threadIdx.x & 31;
    int ntiles = NC >> 4;
    int mt = wave / ntiles;
    int nt = wave - mt * ntiles;
    if (mt * 16 >= M) return;                 // wave-uniform guard

    int r16 = lane & 15;
    int kb  = (lane >> 4) << 1;               // 0 or 2
    const float* Arow = A + (size_t)(mt * 16 + r16) * K;
    const float* Bcol = B + (nt * 16 + r16);

    v8f acc = {};
    int ksfull = K & ~3;
    int ks = 0;
    for (; ks < ksfull; ks += 4) {
        int k0 = ks + kb;
        v2f a, b;
        a.x = Arow[k0];
        a.y = Arow[k0 + 1];
        b.x = Bcol[(size_t)k0 * NC];
        b.y = Bcol[(size_t)(k0 + 1) * NC];
        acc = __builtin_amdgcn_wmma_f32_16x16x4_f32(false, a, false, b,
                                                    (short)0, acc, false, false);
    }
    if (ks < K) {                             // guarded tail (K=22)
        int k0 = ks + kb;
        int k1 = k0 + 1;
        v2f a, b;
        a.x = (k0 < K) ? Arow[k0] : 0.0f;
        a.y = (k1 < K) ? Arow[k1] : 0.0f;
        b.x = (k0 < K) ? Bcol[(size_t)k0 * NC] : 0.0f;
        b.y = (k1 < K) ? Bcol[(size_t)k1 * NC] : 0.0f;
        acc = __builtin_amdgcn_wmma_f32_16x16x4_f32(false, a, false, b,
                                                    (short)0, acc, false, false);
    }

    int orow = mt * 16 + ((lane >> 4) << 3);
    int ocol = nt * 16 + r16;
#pragma unroll
    for (int r = 0; r < 8; ++r)
        Cout[(size_t)(orow + r) * NC + ocol] = acc[r];
}

// ---------------------------------------------------------------------------
// SPMM scatter: out[dst[e]] += w[e] * feat[src[e]].  CH = F/4 float4 chunks
// per edge; consecutive threads cover consecutive chunks of the same edge so
// the gather is fully coalesced. Hardware FP32 atomics into L2.
// ---------------------------------------------------------------------------
template<int LOGCH>
__global__ void spmm_scatter(const float4* __restrict__ feat,
                             const int* __restrict__ esrc,
                             const int* __restrict__ edst,
                             const float* __restrict__ ew,
                             float* __restrict__ out) {
    constexpr int CH = 1 << LOGCH;
    long long t = (long long)blockIdx.x * blockDim.x + threadIdx.x;
    int e  = (int)(t >> LOGCH);
    int ch = (int)(t & (CH - 1));
    if (e >= GE) return;
    int s = esrc[e];
    int d = edst[e];
    float w = ew[e];
    float4 v = feat[(size_t)s * CH + ch];
    float* dst = out + ((size_t)d << (LOGCH + 2)) + (ch << 2);
    unsafeAtomicAdd(dst + 0, v.x * w);
    unsafeAtomicAdd(dst + 1, v.y * w);
    unsafeAtomicAdd(dst + 2, v.z * w);
    unsafeAtomicAdd(dst + 3, v.w * w);
}

// ---------------------------------------------------------------------------
// Elementwise bias (+ optional ReLU), in place. F is a power of two.
// ---------------------------------------------------------------------------
template<int RELU>
__global__ void bias_act(float* __restrict__ buf, const float* __restrict__ bias,
                         int total, int fmask) {
    int i = blockIdx.x * blockDim.x + threadIdx.x;
    if (i >= total) return;
    float v = buf[i] + bias[i & fmask];
    if (RELU) v = fmaxf(v, 0.0f);
    buf[i] = v;
}

// ---------------------------------------------------------------------------
// Head: logits = h @ Wm + bm, then log_softmax. One wave32 per node; lane
// covers 8 consecutive features; Wm row (4 classes) loaded as float4.
// ---------------------------------------------------------------------------
__global__ void head_logsoftmax(const float* __restrict__ h,
                                const float* __restrict__ Wm,
                                const float* __restrict__ bm,
                                float* __restrict__ out) {
    int wave = (int)((blockIdx.x * (long long)blockDim.x + threadIdx.x) >> 5);
    int lane = threadIdx.x & 31;
    if (wave >= GN) return;
    const float* row = h + (size_t)wave * H2F;
    const float4* wm4 = (const float4*)Wm;

    float a0 = 0.f, a1 = 0.f, a2 = 0.f, a3 = 0.f;
#pragma unroll
    for (int i = 0; i < 8; ++i) {
        int f = lane * 8 + i;
        float v = row[f];
        float4 wr = wm4[f];
        a0 += v * wr.x; a1 += v * wr.y; a2 += v * wr.z; a3 += v * wr.w;
    }
#pragma unroll
    for (int off = 16; off >= 1; off >>= 1) {
        a0 += __shfl_xor(a0, off, 32);
        a1 += __shfl_xor(a1, off, 32);
        a2 += __shfl_xor(a2, off, 32);
        a3 += __shfl_xor(a3, off, 32);
    }
    if (lane == 0) {
        float z0 = a0 + bm[0], z1 = a1 + bm[1], z2 = a2 + bm[2], z3 = a3 + bm[3];
        float m  = fmaxf(fmaxf(z0, z1), fmaxf(z2, z3));
        float s  = expf(z0 - m) + expf(z1 - m) + expf(z2 - m) + expf(z3 - m);
        float ls = logf(s);
        float* o = out + (size_t)wave * NCLS;
        o[0] = z0 - m - ls; o[1] = z1 - m - ls; o[2] = z2 - m - ls; o[3] = z3 - m - ls;
    }
}

// ---------------------------------------------------------------------------
extern "C" void kernel_launch(void* const* d_in, const int* in_sizes, int n_in,
                              void* d_out, int out_size, void* d_ws, size_t ws_size,
                              hipStream_t stream) {
    const float* x    = (const float*)d_in[0];
    const int*   esrc = (const int*)d_in[1];
    const int*   edst = (const int*)d_in[2];
    const float* ew   = (const float*)d_in[3];
    const float* W1   = (const float*)d_in[4];
    const float* b1   = (const float*)d_in[5];
    const float* W2   = (const float*)d_in[6];
    const float* b2   = (const float*)d_in[7];
    const float* Wm   = (const float*)d_in[8];
    const float* bm   = (const float*)d_in[9];
    float* out = (float*)d_out;

    // Workspace layout (floats): [xw1: N*128][h1: N*128][hw2: N*256]
    // h2 (N*256) reuses the xw1+h1 region once they are dead.
    float* xw1 = (float*)d_ws;
    float* h1  = xw1 + (size_t)GN * H1F;
    float* hw2 = h1  + (size_t)GN * H1F;
    float* h2  = (float*)d_ws;

    // 1) XW1 = X @ W1            (waves = 6250 * 8 = 50000)
    {
        int waves = (GN / 16) * (H1F / 16);
        gemm_wmma_f32<<<dim3(waves * 32 / 256), dim3(256), 0, stream>>>(
            x, W1, xw1, GN, F_IN, H1F);
    }
    // 2) h1 = A @ XW1 ; h1 = relu(h1 + b1)
    hipMemsetAsync(h1, 0, (size_t)GN * H1F * sizeof(float), stream);
    {
        long long thr = (long long)GE * (H1F / 4);          // 51.2M
        spmm_scatter<5><<<dim3((unsigned)(thr / 256)), dim3(256), 0, stream>>>(
            (const float4*)xw1, esrc, edst, ew, h1);
        int total = GN * H1F;
        bias_act<1><<<dim3(total / 256), dim3(256), 0, stream>>>(h1, b1, total, H1F - 1);
    }
    // 3) HW2 = h1 @ W2           (waves = 6250 * 16 = 100000)
    {
        int waves = (GN / 16) * (H2F / 16);
        gemm_wmma_f32<<<dim3(waves * 32 / 256), dim3(256), 0, stream>>>(
            h1, W2, hw2, GN, H1F, H2F);
    }
    // 4) h2 = A @ HW2 ; h2 += b2   (h2 overlays xw1+h1, now dead)
    hipMemsetAsync(h2, 0, (size_t)GN * H2F * sizeof(float), stream);
    {
        long long thr = (long long)GE * (H2F / 4);          // 102.4M
        spmm_scatter<6><<<dim3((unsigned)(thr / 256)), dim3(256), 0, stream>>>(
            (const float4*)hw2, esrc, edst, ew, h2);
        int total = GN * H2F;
        bias_act<0><<<dim3(total / 256), dim3(256), 0, stream>>>(h2, b2, total, H2F - 1);
    }
    // 5) out = log_softmax(h2 @ Wm + bm)   (one wave per node)
    {
        int waves = GN;                                      // 100000 waves
        int blocks = (waves * 32 + 255) / 256;               // 12500
        head_logsoftmax<<<dim3(blocks), dim3(256), 0, stream>>>(h2, Wm, bm, out);
    }
}